// SANLayer_55645596287748
// MI455X (gfx1250) — compile-verified
//
#include <hip/hip_runtime.h>
#include <cmath>

// ---------------------------------------------------------------------------
// SAN layer for MI455X (gfx1250, wave32).
//
//   out = att_dn @ (z_dn0 + att_dn @ z_dn1)
//       + att_up @ (z_up0 + att_up @ z_up1)
//       + proj   @ (x @ W_har)
//
// using (att@att)@z == att@(att@z). All heavy matmuls run on
// V_WMMA_F32_16X16X4_F32 (fp32-exact vs reference). Attention weights are
// computed on-the-fly from the 256MB Laplacians (elu + online-softmax stats),
// staged through LDS — bandwidth-bound at ~1.8 GB total (~77us @ 23.3 TB/s).
// ---------------------------------------------------------------------------

typedef __attribute__((ext_vector_type(2))) float v2f;
typedef __attribute__((ext_vector_type(4))) float v4f;
typedef __attribute__((ext_vector_type(8))) float v8f;

#define N_NODES 8192
#define DIN     128
#define DOUT    128
#define KSPLIT  4
#define BK      64

__device__ __forceinline__ float eluf(float x) {
  return x > 0.f ? x : expm1f(x);
}

__device__ __forceinline__ v8f wmma_f32(v2f a, v2f b, v8f c) {
  // (neg_a, A, neg_b, B, c_mod, C, reuse_a, reuse_b)
  return __builtin_amdgcn_wmma_f32_16x16x4_f32(false, a, false, b, (short)0, c,
                                               false, false);
}

// ---------------------------------------------------------------------------
// Kernel 1: the five small GEMMs  Z[m] = x @ W_m   (8192x128 @ 128x128)
// One wave per 16x16 output tile; K=128 -> 32 WMMA_F32_16X16X4 per wave.
// Z slot order: [up0, up1, dn0, dn1, har]
// ---------------------------------------------------------------------------
__global__ __launch_bounds__(256)
void xw_gemm_kernel(const float* __restrict__ x,
                    const float* __restrict__ W_up,
                    const float* __restrict__ W_dn,
                    const float* __restrict__ W_har,
                    float* __restrict__ Z)
{
  const int lane = threadIdx.x & 31;
  const int wave = threadIdx.x >> 5;
  const int tile = blockIdx.x * 8 + wave;        // 20480 tiles total
  const int mat  = tile >> 12;                   // /4096
  const int rem  = tile & 4095;
  const int rt   = rem >> 3;
  const int ct   = rem & 7;

  const float* Wm;
  switch (mat) {
    case 0:  Wm = W_up;               break;
    case 1:  Wm = W_up + DIN * DOUT;  break;
    case 2:  Wm = W_dn;               break;
    case 3:  Wm = W_dn + DIN * DOUT;  break;
    default: Wm = W_har;              break;
  }

  const int row0 = rt * 16, col0 = ct * 16;
  const int mn   = lane & 15;          // A: M index / B: N index
  const int koff = (lane >> 4) * 2;    // K sub-offset per half-wave

  v8f acc = {};
  #pragma unroll 4
  for (int k = 0; k < DIN; k += 4) {
    v2f a = *(const v2f*)(x + (size_t)(row0 + mn) * DIN + k + koff);
    v2f b;
    b.x = Wm[(k + koff    ) * DOUT + col0 + mn];
    b.y = Wm[(k + koff + 1) * DOUT + col0 + mn];
    acc = wmma_f32(a, b, acc);
  }

  float* Zo = Z + (size_t)mat * N_NODES * DOUT;
  #pragma unroll
  for (int j = 0; j < 8; ++j) {
    const int r = row0 + j + (lane >> 4) * 8;
    Zo[(size_t)r * DOUT + col0 + (lane & 15)] = acc[j];
  }
}

// ---------------------------------------------------------------------------
// Kernel 2: s,t vectors.  s_n = sum_{f,o} z[f][n][o] * a_src[f*128+o], etc.
// vecs layout: [s_dn | t_dn | s_up | t_up], 8192 floats each.
// ---------------------------------------------------------------------------
__global__ __launch_bounds__(256)
void st_kernel(const float* __restrict__ Z,
               const float* __restrict__ a_up,
               const float* __restrict__ a_dn,
               float* __restrict__ vecs)
{
  const int idx = blockIdx.x * blockDim.x + threadIdx.x;
  if (idx >= 2 * N_NODES) return;
  const int c = idx >> 13;                 // 0 = down, 1 = up
  const int n = idx & (N_NODES - 1);
  const float* a  = c ? a_up : a_dn;
  const float* z0 = Z + (size_t)(c ? 0 : 2) * N_NODES * DOUT + (size_t)n * DOUT;
  const float* z1 = z0 + (size_t)N_NODES * DOUT;
  float s = 0.f, t = 0.f;
  #pragma unroll 4
  for (int o = 0; o < DOUT; ++o) {
    const float v0 = z0[o], v1 = z1[o];
    s += v0 * a[o]            + v1 * a[DOUT + o];
    t += v0 * a[2 * DOUT + o] + v1 * a[3 * DOUT + o];
  }
  vecs[c * 2 * N_NODES + n]           = s;
  vecs[c * 2 * N_NODES + N_NODES + n] = t;
}

// ---------------------------------------------------------------------------
// Kernel 3: masked online-softmax row stats over the Laplacian.
// One 256-thread block per (row, conv). stats: [m_dn | il_dn | m_up | il_up].
// ---------------------------------------------------------------------------
__global__ __launch_bounds__(256)
void row_stats_kernel(const float* __restrict__ lap_dn,
                      const float* __restrict__ lap_up,
                      const float* __restrict__ vecs,
                      float* __restrict__ stats)
{
  __shared__ float sm[256], sl[256];
  const int i   = blockIdx.x;
  const int c   = blockIdx.y;
  const int tid = threadIdx.x;
  const float* lap = c ? lap_up : lap_dn;
  const float* sv  = vecs + c * 2 * N_NODES;
  const float  ti  = vecs[c * 2 * N_NODES + N_NODES + i];
  const float* row = lap + (size_t)i * N_NODES;

  float m = -__builtin_inff(), l = 0.f;
  for (int j = tid; j < N_NODES; j += 256) {
    const float v = row[j];
    if (v != 0.f) {
      const float e = eluf(ti + sv[j]);
      if (e <= m) {
        l += __expf(e - m);
      } else {
        l = (l > 0.f ? l * __expf(m - e) : 0.f) + 1.f;
        m = e;
      }
    }
  }
  sm[tid] = m; sl[tid] = l;
  __syncthreads();
  for (int s = 128; s > 0; s >>= 1) {
    if (tid < s) {
      const float m1 = sm[tid], l1 = sl[tid];
      const float m2 = sm[tid + s], l2 = sl[tid + s];
      const float M = fmaxf(m1, m2);
      const float L = (l1 > 0.f ? l1 * __expf(m1 - M) : 0.f)
                    + (l2 > 0.f ? l2 * __expf(m2 - M) : 0.f);
      sm[tid] = M; sl[tid] = L;
    }
    __syncthreads();
  }
  if (tid == 0) {
    stats[c * 2 * N_NODES + i]           = sm[0];
    stats[c * 2 * N_NODES + N_NODES + i] = (sl[0] > 0.f) ? 1.f / sl[0] : 0.f;
  }
}

// ---------------------------------------------------------------------------
// Kernel 4: fused  dest += f(P) @ V  where f is either identity (proj) or the
// on-the-fly attention transform of the Laplacian.
// Block = 8 waves; each block owns a 16-row strip x all 128 cols; wave w owns
// the 16-col tile at 16*w. K split KSPLIT ways across blockIdx.y; partial
// results accumulate via global_atomic_add_f32.
// A-tile (16xBK) staged through LDS (stride BK+4 -> conflict-free float2
// reads in the WMMA A layout); B read straight from L2-resident V (4 MB).
// ---------------------------------------------------------------------------
__global__ __launch_bounds__(256)
void att_spmm_kernel(const float* __restrict__ P,
                     const float* __restrict__ sv,
                     const float* __restrict__ tv,
                     const float* __restrict__ mr,
                     const float* __restrict__ il,
                     const float* __restrict__ V,
                     float* __restrict__ dest,
                     int att_mode)
{
  __shared__ float sP[16][BK + 4];
  const int tid  = threadIdx.x;
  const int lane = tid & 31;
  const int wave = tid >> 5;
  const int row0 = blockIdx.x * 16;
  const int kbeg = blockIdx.y * (N_NODES / KSPLIT);
  const int kend = kbeg + (N_NODES / KSPLIT);

  // staging coordinates: 16 rows x 64 cols, float4 per thread, coalesced
  const int sr   = tid >> 4;         // 0..15
  const int sc   = (tid & 15) * 4;   // 0..60
  const int grow = row0 + sr;
  float row_t = 0.f, row_m = 0.f, row_il = 0.f;
  if (att_mode) { row_t = tv[grow]; row_m = mr[grow]; row_il = il[grow]; }
  const float* Prow = P + (size_t)grow * N_NODES;

  const int col0 = wave * 16;
  const int mn   = lane & 15;
  const int koff = (lane >> 4) * 2;
  v8f acc = {};

  for (int kk = kbeg; kk < kend; kk += BK) {
    // ---- stage + transform the P tile into LDS ----
    v4f p = *(const v4f*)(Prow + kk + sc);
    if (kk + BK < kend)
      __builtin_prefetch(Prow + kk + BK + sc, 0, 1);  // global_prefetch next tile

    float o0, o1, o2, o3;
    if (att_mode) {
      const float* sj = sv + kk + sc;
      o0 = (p.x != 0.f) ? __expf(eluf(row_t + sj[0]) - row_m) * row_il : 0.f;
      o1 = (p.y != 0.f) ? __expf(eluf(row_t + sj[1]) - row_m) * row_il : 0.f;
      o2 = (p.z != 0.f) ? __expf(eluf(row_t + sj[2]) - row_m) * row_il : 0.f;
      o3 = (p.w != 0.f) ? __expf(eluf(row_t + sj[3]) - row_m) * row_il : 0.f;
    } else {
      o0 = p.x; o1 = p.y; o2 = p.z; o3 = p.w;
    }
    sP[sr][sc + 0] = o0;
    sP[sr][sc + 1] = o1;
    sP[sr][sc + 2] = o2;
    sP[sr][sc + 3] = o3;
    __syncthreads();

    // ---- 16 WMMA K-steps over the staged tile ----
    #pragma unroll 4
    for (int k0 = 0; k0 < BK; k0 += 4) {
      v2f a = *(const v2f*)&sP[mn][k0 + koff];
      const float* vp = V + (size_t)(kk + k0 + koff) * DOUT + col0 + mn;
      v2f b;
      b.x = vp[0];
      b.y = vp[DOUT];
      acc = wmma_f32(a, b, acc);
    }
    __syncthreads();
  }

  #pragma unroll
  for (int j = 0; j < 8; ++j) {
    const int r = row0 + j + (lane >> 4) * 8;
    atomicAdd(dest + (size_t)r * DOUT + col0 + (lane & 15), acc[j]);
  }
}

// ---------------------------------------------------------------------------
// Host orchestration (graph-capture safe: only async ops on `stream`).
// ---------------------------------------------------------------------------
extern "C" void kernel_launch(void* const* d_in, const int* in_sizes, int n_in,
                              void* d_out, int out_size, void* d_ws, size_t ws_size,
                              hipStream_t stream)
{
  (void)in_sizes; (void)n_in; (void)ws_size;

  const float* x      = (const float*)d_in[0];
  const float* lap_up = (const float*)d_in[1];
  const float* lap_dn = (const float*)d_in[2];
  const float* proj   = (const float*)d_in[3];
  const float* W_up   = (const float*)d_in[4];
  const float* a_up   = (const float*)d_in[5];
  const float* W_dn   = (const float*)d_in[6];
  const float* a_dn   = (const float*)d_in[7];
  const float* W_har  = (const float*)d_in[8];
  float* out = (float*)d_out;
  float* ws  = (float*)d_ws;

  // workspace layout (floats)
  float* Z     = ws;                                   // 5 * 8192*128
  float* U     = Z + 5ull * N_NODES * DOUT;            // 8192*128
  float* vecs  = U + (size_t)N_NODES * DOUT;           // s/t: 4 * 8192
  float* stats = vecs + 4 * N_NODES;                   // m/invl: 4 * 8192

  const float* z_up0 = Z;
  const float* z_up1 = Z + 1ull * N_NODES * DOUT;
  const float* z_dn0 = Z + 2ull * N_NODES * DOUT;
  const float* z_dn1 = Z + 3ull * N_NODES * DOUT;
  const float* xwhar = Z + 4ull * N_NODES * DOUT;
  const float* s_dn = vecs,               * t_dn = vecs + N_NODES;
  const float* s_up = vecs + 2 * N_NODES, * t_up = vecs + 3 * N_NODES;
  const float* m_dn = stats,               * il_dn = stats + N_NODES;
  const float* m_up = stats + 2 * N_NODES, * il_up = stats + 3 * N_NODES;

  const size_t zbytes = (size_t)N_NODES * DOUT * sizeof(float);

  hipMemsetAsync(out, 0, (size_t)out_size * sizeof(float), stream);

  xw_gemm_kernel<<<dim3(2560), dim3(256), 0, stream>>>(x, W_up, W_dn, W_har, Z);
  st_kernel<<<dim3(64), dim3(256), 0, stream>>>(Z, a_up, a_dn, vecs);
  row_stats_kernel<<<dim3(N_NODES, 2), dim3(256), 0, stream>>>(lap_dn, lap_up,
                                                               vecs, stats);

  const dim3 g(N_NODES / 16, KSPLIT), b(256);

  // conv down:  U = z_dn0 + att_dn @ z_dn1 ;  out += att_dn @ U
  hipMemcpyAsync(U, z_dn0, zbytes, hipMemcpyDeviceToDevice, stream);
  att_spmm_kernel<<<g, b, 0, stream>>>(lap_dn, s_dn, t_dn, m_dn, il_dn,
                                       z_dn1, U, 1);
  att_spmm_kernel<<<g, b, 0, stream>>>(lap_dn, s_dn, t_dn, m_dn, il_dn,
                                       U, out, 1);

  // conv up:    U = z_up0 + att_up @ z_up1 ;  out += att_up @ U
  hipMemcpyAsync(U, z_up0, zbytes, hipMemcpyDeviceToDevice, stream);
  att_spmm_kernel<<<g, b, 0, stream>>>(lap_up, s_up, t_up, m_up, il_up,
                                       z_up1, U, 1);
  att_spmm_kernel<<<g, b, 0, stream>>>(lap_up, s_up, t_up, m_up, il_up,
                                       U, out, 1);

  // harmonic:   out += proj @ (x @ W_har)
  att_spmm_kernel<<<g, b, 0, stream>>>(proj, nullptr, nullptr, nullptr, nullptr,
                                       xwhar, out, 0);
}